// HGTLayer_49134425866254
// MI455X (gfx1250) — compile-verified
//
#include <hip/hip_runtime.h>
#include <hip/hip_bf16.h>

#define NNODES 100000
#define NEDGES 320000
#define DIM    256
#define HEADS  8
#define DHEAD  32
#define SCALE  0.17677669529663687f  // 32^-0.5

typedef float v2f __attribute__((ext_vector_type(2)));
typedef float v8f __attribute__((ext_vector_type(8)));
typedef int   v4i __attribute__((ext_vector_type(4)));

#if defined(__has_builtin)
#if __has_builtin(__builtin_amdgcn_global_load_async_to_lds_b128) && \
    __has_builtin(__builtin_amdgcn_s_wait_asynccnt)
#define USE_ASYNC_LDS 1
#endif
#endif

// order-preserving float -> uint key for atomic max
__device__ __forceinline__ unsigned fkey(float f) {
  unsigned u = __float_as_uint(f);
  return (u & 0x80000000u) ? ~u : (u | 0x80000000u);
}
__device__ __forceinline__ float funkey(unsigned u) {
  return __uint_as_float((u & 0x80000000u) ? (u ^ 0x80000000u) : ~u);
}

__global__ void fill_u32_kernel(unsigned* __restrict__ p, unsigned v, long long n) {
  long long i = (long long)blockIdx.x * blockDim.x + threadIdx.x;
  if (i < n) p[i] = v;
}

// C[M,256] (+)= A[M,256] @ B[256,256], fp32 WMMA 16x16x4.
// Block = 4 waves. Block handles 4 row-tiles (one per wave) x 64 cols.
// The 256x64 B panel is staged once per block in LDS (async DMA to LDS when
// available). Grid is linearized with the column panel varying fastest so
// consecutive blocks reuse the same A rows out of L2.
__global__ __launch_bounds__(128)
void gemm256_wmma_kernel(const float* __restrict__ A, const float* __restrict__ B,
                         float* __restrict__ C, int mtiles, int accumulate) {
  __shared__ float Bs[256 * 64];  // 64 KB panel: B[:, nbase:nbase+64]

  const int tid  = threadIdx.x;          // 0..127
  const int wave = tid >> 5;
  const int lane = tid & 31;
  const int lin  = blockIdx.x;
  const int nbase = (lin & 3) * 64;      // column panel (fastest)
  const int mt    = (lin >> 2) * 4 + wave;

  // ---- stage B panel into LDS: 4096 x 16B chunks, 32 per thread ----
  const float* Bg = B + nbase;
  #pragma unroll
  for (int i = 0; i < 32; ++i) {
    int c    = tid + (i << 7);      // chunk id 0..4095
    int row  = c >> 4;              // 0..255
    int col4 = (c & 15) << 2;       // 0,4,...,60
    const float* gp = Bg + row * DIM + col4;
    float*       lp = &Bs[row * 64 + col4];
#if USE_ASYNC_LDS
    __builtin_amdgcn_global_load_async_to_lds_b128((v4i*)gp, (v4i*)lp, 0, 0);
#else
    *(float4*)lp = *(const float4*)gp;
#endif
  }
#if USE_ASYNC_LDS
  __builtin_amdgcn_s_wait_asynccnt(0);
#endif
  __syncthreads();

  if (mt < mtiles) {
    const int mbase = mt * 16;
    const int half  = lane >> 4;    // 0: K=0,1   1: K=2,3
    const int l16   = lane & 15;

    v8f acc0 = {}, acc1 = {}, acc2 = {}, acc3 = {};
    const float* arow = A + (long long)(mbase + l16) * DIM;

    for (int k = 0; k < DIM; k += 4) {
      v2f a;
      a.x = arow[k + half * 2 + 0];
      a.y = arow[k + half * 2 + 1];
      const float* bp = &Bs[(k + half * 2) * 64 + l16];  // bp[64] = next K row
      v2f bb;
      bb.x = bp[0];   bb.y = bp[64];
      acc0 = __builtin_amdgcn_wmma_f32_16x16x4_f32(false, a, false, bb, (short)0, acc0, false, false);
      bb.x = bp[16];  bb.y = bp[80];
      acc1 = __builtin_amdgcn_wmma_f32_16x16x4_f32(false, a, false, bb, (short)0, acc1, false, false);
      bb.x = bp[32];  bb.y = bp[96];
      acc2 = __builtin_amdgcn_wmma_f32_16x16x4_f32(false, a, false, bb, (short)0, acc2, false, false);
      bb.x = bp[48];  bb.y = bp[112];
      acc3 = __builtin_amdgcn_wmma_f32_16x16x4_f32(false, a, false, bb, (short)0, acc3, false, false);
    }

    // C/D layout: VGPR r holds row mbase + r + 8*half, col nbase + l16 (+16*tile)
    #pragma unroll
    for (int r = 0; r < 8; ++r) {
      int row = mbase + r + half * 8;
      float* crow = C + (long long)row * DIM + nbase + l16;
      if (accumulate) {
        crow[0]  += acc0[r]; crow[16] += acc1[r];
        crow[32] += acc2[r]; crow[48] += acc3[r];
      } else {
        crow[0]  = acc0[r]; crow[16] = acc1[r];
        crow[32] = acc2[r]; crow[48] = acc3[r];
      }
    }
  }
}

// Pass 1: per-(edge,head) attention score + running segment max (atomic on uint key)
__global__ void edge_score_kernel(const float* __restrict__ Qd, const float* __restrict__ Ks,
                                  const int* __restrict__ src, const int* __restrict__ dst,
                                  const float* __restrict__ mu,
                                  float* __restrict__ attn, unsigned* __restrict__ mkey,
                                  int nedges) {
  int gid = blockIdx.x * blockDim.x + threadIdx.x;
  if (gid >= nedges * HEADS) return;
  int e = gid >> 3;
  int h = gid & 7;
  int s = src[e];
  int d = dst[e];
  const float* q = Qd + (long long)d * DIM + h * DHEAD;
  const float* k = Ks + (long long)s * DIM + h * DHEAD;
  float dot = 0.f;
  #pragma unroll
  for (int i = 0; i < DHEAD; i += 4) {
    float4 qv = *(const float4*)(q + i);
    float4 kv = *(const float4*)(k + i);
    dot = fmaf(qv.x, kv.x, dot);
    dot = fmaf(qv.y, kv.y, dot);
    dot = fmaf(qv.z, kv.z, dot);
    dot = fmaf(qv.w, kv.w, dot);
  }
  float sig = 1.0f / (1.0f + expf(-mu[h]));
  float a = dot * SCALE * sig;
  attn[gid] = a;
  atomicMax(mkey + (long long)d * HEADS + h, fkey(a));
}

// Pass 2: e = exp(attn - max), accumulate denominators
__global__ void edge_exp_kernel(float* __restrict__ attn, const int* __restrict__ dst,
                                const unsigned* __restrict__ mkey, float* __restrict__ denom,
                                int nedges) {
  int gid = blockIdx.x * blockDim.x + threadIdx.x;
  if (gid >= nedges * HEADS) return;
  int e = gid >> 3;
  int h = gid & 7;
  int d = dst[e];
  float m = funkey(mkey[(long long)d * HEADS + h]);
  float ev = expf(attn[gid] - m);
  attn[gid] = ev;
  atomicAdd(denom + (long long)d * HEADS + h, ev);
}

// Pass 3: wv[dst, c] += (e/denom) * V[src, c]
__global__ void edge_accum_kernel(const float* __restrict__ ew, const float* __restrict__ denom,
                                  const float* __restrict__ Vs,
                                  const int* __restrict__ src, const int* __restrict__ dst,
                                  float* __restrict__ wv) {
  long long gid = (long long)blockIdx.x * blockDim.x + threadIdx.x;
  if (gid >= (long long)NEDGES * DIM) return;
  int e = (int)(gid >> 8);
  int c = (int)(gid & 255);
  int h = c >> 5;
  int s = src[e];
  int d = dst[e];
  float a = ew[(long long)e * HEADS + h] / denom[(long long)d * HEADS + h];
  atomicAdd(wv + (long long)d * DIM + c, a * Vs[(long long)s * DIM + c]);
}

// out = layernorm(relu(agg + bias)) * g + b ; one wave32 per row
__global__ __launch_bounds__(256)
void finalize_kernel(const float* __restrict__ agg, const float* __restrict__ bias,
                     const float* __restrict__ g, const float* __restrict__ b,
                     float* __restrict__ out, int nrows) {
  int wave = threadIdx.x >> 5;
  int lane = threadIdx.x & 31;
  int row = blockIdx.x * 8 + wave;
  if (row >= nrows) return;
  const float* x = agg + (long long)row * DIM;
  float v[8];
  float s = 0.f;
  #pragma unroll
  for (int j = 0; j < 8; ++j) {
    int c = lane + j * 32;
    float t = x[c] + bias[c];
    t = fmaxf(t, 0.f);
    v[j] = t;
    s += t;
  }
  #pragma unroll
  for (int off = 16; off > 0; off >>= 1) s += __shfl_xor(s, off, 32);
  float mean = s * (1.0f / 256.0f);
  float vs = 0.f;
  #pragma unroll
  for (int j = 0; j < 8; ++j) {
    float d = v[j] - mean;
    vs += d * d;
  }
  #pragma unroll
  for (int off = 16; off > 0; off >>= 1) vs += __shfl_xor(vs, off, 32);
  float rstd = rsqrtf(vs * (1.0f / 256.0f) + 1e-5f);
  float* o = out + (long long)row * DIM;
  #pragma unroll
  for (int j = 0; j < 8; ++j) {
    int c = lane + j * 32;
    o[c] = (v[j] - mean) * rstd * g[c] + b[c];
  }
}

extern "C" void kernel_launch(void* const* d_in, const int* in_sizes, int n_in,
                              void* d_out, int out_size, void* d_ws, size_t ws_size,
                              hipStream_t stream) {
  const float* x_a       = (const float*)d_in[0];
  const float* x_b       = (const float*)d_in[1];
  const float* Wq_a      = (const float*)d_in[2];
  const float* Wk_a      = (const float*)d_in[3];
  const float* Wv_a      = (const float*)d_in[4];
  const float* Wq_b      = (const float*)d_in[5];
  const float* Wk_b      = (const float*)d_in[6];
  const float* Wv_b      = (const float*)d_in[7];
  const float* Wskip_a_w = (const float*)d_in[8];
  const float* Wskip_a_b = (const float*)d_in[9];
  const float* Wskip_b_w = (const float*)d_in[10];
  const float* Wskip_b_b = (const float*)d_in[11];
  const float* g_a       = (const float*)d_in[12];
  const float* b_a       = (const float*)d_in[13];
  const float* g_b       = (const float*)d_in[14];
  const float* b_b       = (const float*)d_in[15];
  const float* mu_ab     = (const float*)d_in[16];
  const float* Wmsg_ab   = (const float*)d_in[17];
  const float* mu_ba     = (const float*)d_in[18];
  const float* Wmsg_ba   = (const float*)d_in[19];
  const int*   ei_ab     = (const int*)d_in[20];  // [2, E]: src, dst
  const int*   ei_ba     = (const int*)d_in[21];

  const long long SN = (long long)NNODES * DIM;   // 25.6M floats per matrix
  float* ws = (float*)d_ws;
  float* Qa    = ws + 0 * SN;
  float* Ka    = ws + 1 * SN;
  float* Va    = ws + 2 * SN;
  float* Qb    = ws + 3 * SN;
  float* Kb    = ws + 4 * SN;
  float* Vb    = ws + 5 * SN;
  float* wv_a  = ws + 6 * SN;
  float* wv_b  = ws + 7 * SN;
  float* agg_a = ws + 8 * SN;
  float* agg_b = ws + 9 * SN;
  float* attn_ab = ws + 10 * SN;
  float* attn_ba = attn_ab + (long long)NEDGES * HEADS;
  unsigned* mkey_b  = (unsigned*)(attn_ba + (long long)NEDGES * HEADS);
  unsigned* mkey_a  = mkey_b + (long long)NNODES * HEADS;
  float* denom_b = (float*)(mkey_a + (long long)NNODES * HEADS);
  float* denom_a = denom_b + (long long)NNODES * HEADS;

  // --- init: zero wv_a/wv_b and mkey/denom (contiguous: mkey_b..denom_a end)
  {
    long long n = 2 * SN;
    fill_u32_kernel<<<(unsigned)((n + 255) / 256), 256, 0, stream>>>((unsigned*)wv_a, 0u, n);
    long long n2 = 4LL * NNODES * HEADS;
    fill_u32_kernel<<<(unsigned)((n2 + 255) / 256), 256, 0, stream>>>(mkey_b, 0u, n2);
  }

  // --- Q/K/V projections (fp32 WMMA GEMMs, LDS-staged B panels)
  const int mtiles = NNODES / 16;  // 6250
  dim3 ggrid((unsigned)(((mtiles + 3) / 4) * 4));  // col panel varies fastest
  dim3 gblock(128);
  gemm256_wmma_kernel<<<ggrid, gblock, 0, stream>>>(x_a, Wq_a, Qa, mtiles, 0);
  gemm256_wmma_kernel<<<ggrid, gblock, 0, stream>>>(x_a, Wk_a, Ka, mtiles, 0);
  gemm256_wmma_kernel<<<ggrid, gblock, 0, stream>>>(x_a, Wv_a, Va, mtiles, 0);
  gemm256_wmma_kernel<<<ggrid, gblock, 0, stream>>>(x_b, Wq_b, Qb, mtiles, 0);
  gemm256_wmma_kernel<<<ggrid, gblock, 0, stream>>>(x_b, Wk_b, Kb, mtiles, 0);
  gemm256_wmma_kernel<<<ggrid, gblock, 0, stream>>>(x_b, Wv_b, Vb, mtiles, 0);

  // --- edge attention (relation ab: src=a, dst=b ; relation ba: src=b, dst=a)
  const int nEH = NEDGES * HEADS;
  const int ebl = 256;
  edge_score_kernel<<<(nEH + ebl - 1) / ebl, ebl, 0, stream>>>(
      Qb, Ka, ei_ab, ei_ab + NEDGES, mu_ab, attn_ab, mkey_b, NEDGES);
  edge_score_kernel<<<(nEH + ebl - 1) / ebl, ebl, 0, stream>>>(
      Qa, Kb, ei_ba, ei_ba + NEDGES, mu_ba, attn_ba, mkey_a, NEDGES);

  edge_exp_kernel<<<(nEH + ebl - 1) / ebl, ebl, 0, stream>>>(
      attn_ab, ei_ab + NEDGES, mkey_b, denom_b, NEDGES);
  edge_exp_kernel<<<(nEH + ebl - 1) / ebl, ebl, 0, stream>>>(
      attn_ba, ei_ba + NEDGES, mkey_a, denom_a, NEDGES);

  {
    long long nED = (long long)NEDGES * DIM;
    unsigned blocks = (unsigned)((nED + ebl - 1) / ebl);
    edge_accum_kernel<<<blocks, ebl, 0, stream>>>(attn_ab, denom_b, Va, ei_ab, ei_ab + NEDGES, wv_b);
    edge_accum_kernel<<<blocks, ebl, 0, stream>>>(attn_ba, denom_a, Vb, ei_ba, ei_ba + NEDGES, wv_a);
  }

  // --- message mixing GEMM (distributed over segment sum) + skip accumulate
  gemm256_wmma_kernel<<<ggrid, gblock, 0, stream>>>(wv_a, Wmsg_ba, agg_a, mtiles, 0);
  gemm256_wmma_kernel<<<ggrid, gblock, 0, stream>>>(wv_b, Wmsg_ab, agg_b, mtiles, 0);
  gemm256_wmma_kernel<<<ggrid, gblock, 0, stream>>>(x_a, Wskip_a_w, agg_a, mtiles, 1);
  gemm256_wmma_kernel<<<ggrid, gblock, 0, stream>>>(x_b, Wskip_b_w, agg_b, mtiles, 1);

  // --- bias + relu + layernorm -> outputs (out_a then out_b, concatenated)
  float* out_a = (float*)d_out;
  float* out_b = out_a + SN;
  finalize_kernel<<<(NNODES + 7) / 8, 256, 0, stream>>>(agg_a, Wskip_a_b, g_a, b_a, out_a, NNODES);
  finalize_kernel<<<(NNODES + 7) / 8, 256, 0, stream>>>(agg_b, Wskip_b_b, g_b, b_b, out_b, NNODES);
}